// ResidualModule_17609365914197
// MI455X (gfx1250) — compile-verified
//
#include <hip/hip_runtime.h>

typedef __attribute__((ext_vector_type(2))) float v2f;
typedef __attribute__((ext_vector_type(8))) float v8f;

#define CH 128  // both drug and protein channel counts

// ---------------------------------------------------------------------------
// Zero a float4 region (agg accumulators live in poisoned workspace)
// ---------------------------------------------------------------------------
__global__ void zero_f4(float4* __restrict__ p, long n4) {
  long i = (long)blockIdx.x * blockDim.x + threadIdx.x;
  long stride = (long)gridDim.x * blockDim.x;
  float4 z; z.x = 0.f; z.y = 0.f; z.z = 0.f; z.w = 0.f;
  for (; i < n4; i += stride) p[i] = z;
}

// ---------------------------------------------------------------------------
// Edge-wise sum aggregation: accum[dst] += feat[src] (128 channels).
// One wave32 per edge, each lane handles 4 channels: float4 gather +
// 4x global_atomic_add_f32 (relaxed, device scope -> no CAS loop).
// ei layout: ei[0..E) = src, ei[E..2E) = dst (row-major [2,E] int32).
// ---------------------------------------------------------------------------
__global__ __launch_bounds__(256)
void scatter_add(const float* __restrict__ feat,
                 const int* __restrict__ ei,
                 float* __restrict__ accum, int E) {
  long gid = (long)blockIdx.x * blockDim.x + threadIdx.x;
  int e = (int)(gid >> 5);
  int lane = (int)(gid & 31);
  if (e >= E) return;
  int s = ei[e];
  int d = ei[E + e];
  const float4 v = *(const float4*)(feat + (long)s * CH + lane * 4);
  float* dst = accum + (long)d * CH + lane * 4;
  __hip_atomic_fetch_add(dst + 0, v.x, __ATOMIC_RELAXED, __HIP_MEMORY_SCOPE_AGENT);
  __hip_atomic_fetch_add(dst + 1, v.y, __ATOMIC_RELAXED, __HIP_MEMORY_SCOPE_AGENT);
  __hip_atomic_fetch_add(dst + 2, v.z, __ATOMIC_RELAXED, __HIP_MEMORY_SCOPE_AGENT);
  __hip_atomic_fetch_add(dst + 3, v.w, __ATOMIC_RELAXED, __HIP_MEMORY_SCOPE_AGENT);
}

// ---------------------------------------------------------------------------
// One relation's contribution: stage 16x128 A tile in LDS, then K-loop of
// V_WMMA_F32_16X16X4_F32 (32 steps of K=4). Pointers must flow directly
// from kernel args so addrspace inference yields global_load (not flat).
//
// Fragment layouts (CDNA5 ISA 7.12.2, wave32):
//   A 16x4 f32 : lane L holds row m=L&15; vgpr0/1 = K {0,1} for lanes 0-15,
//                K {2,3} for lanes 16-31.
//   B 4x16 f32 : mirrored (vgpr0/1 = K rows, lanes = columns).
//   C/D 16x16  : row = vgpr + 8*(L>=16), col = L&15.
// ---------------------------------------------------------------------------
__device__ __forceinline__
v8f rel_accum(const float* __restrict__ A, const float* __restrict__ W,
              float* As, v8f acc,
              int tid, int nl, int hi, int nbase, int row0, int M) {
  __syncthreads();  // protect LDS reuse across relations
  {
    const float4* src = (const float4*)(A + (long)row0 * CH);
    float4* dst = (float4*)As;
    float4 z; z.x = 0.f; z.y = 0.f; z.z = 0.f; z.w = 0.f;
    #pragma unroll
    for (int i = tid; i < 16 * CH / 4; i += 256) {
      int rr = i >> 5;  // 32 float4 per row
      dst[i] = (row0 + rr < M) ? src[i] : z;
    }
  }
  __syncthreads();

  const float* Wp = W + nbase + nl;  // this wave's column slice of W
  __builtin_prefetch(Wp, 0, 0);      // global_prefetch_b8
  #pragma unroll
  for (int k0 = 0; k0 < CH; k0 += 4) {
    v2f a = *(const v2f*)&As[nl * CH + k0 + 2 * hi];
    v2f b;
    b.x = Wp[(k0 + 2 * hi) * CH];
    b.y = Wp[(k0 + 1 + 2 * hi) * CH];
    acc = __builtin_amdgcn_wmma_f32_16x16x4_f32(
        /*neg_a=*/false, a, /*neg_b=*/false, b,
        /*c_mod=*/(short)0, acc, /*reuse_a=*/false, /*reuse_b=*/false);
  }
  return acc;
}

// ---------------------------------------------------------------------------
// Fused multi-relation node update:
//   out = relu( [residual +] bias + sum_{r<NREL} A_r @ W_r )
// Block = 256 threads = 8 waves; block owns 16 rows, wave w owns columns
// [16w, 16w+16). After the K-loops the 8KB LDS tile (exactly one 16x128
// output tile) is reused to transpose accumulators so the bias/residual/ReLU
// epilogue runs with fully coalesced float4 traffic.
// ---------------------------------------------------------------------------
template<int NREL, bool HAS_RES>
__global__ __launch_bounds__(256)
void gemm_multi(const float* __restrict__ A0, const float* __restrict__ W0,
                const float* __restrict__ A1, const float* __restrict__ W1,
                const float* __restrict__ A2, const float* __restrict__ W2,
                const float* __restrict__ bias,
                const float* __restrict__ residual,
                float* __restrict__ out, int M) {
  __shared__ float As[16 * CH];  // 8 KB: A tile, later output tile

  const int tid   = threadIdx.x;
  const int wave  = tid >> 5;
  const int lane  = tid & 31;
  const int hi    = lane >> 4;   // half-wave select
  const int nl    = lane & 15;   // row (for A) / column (for B,C)
  const int row0  = blockIdx.x * 16;
  const int nbase = wave * 16;

  v8f acc = {};  // C/D accumulator, carried across relations

  acc = rel_accum(A0, W0, As, acc, tid, nl, hi, nbase, row0, M);
  if (NREL >= 2) acc = rel_accum(A1, W1, As, acc, tid, nl, hi, nbase, row0, M);
  if (NREL >= 3) acc = rel_accum(A2, W2, As, acc, tid, nl, hi, nbase, row0, M);

  // Park accumulators in LDS (row-major 16x128 output tile).
  __syncthreads();  // all waves done reading A fragments
  {
    const int col = nbase + nl;
    #pragma unroll
    for (int v = 0; v < 8; ++v) {
      int mrow = v + 8 * hi;  // local row 0..15
      As[mrow * CH + col] = acc[v];
    }
  }
  __syncthreads();

  // Coalesced epilogue: bias (+ residual) + ReLU, float4 stores.
  {
    const float4* t4   = (const float4*)As;
    const float4* b4   = (const float4*)bias;
    const float4* r4   = (const float4*)residual;
    float4*       out4 = (float4*)out;
    #pragma unroll
    for (int i = tid; i < 16 * CH / 4; i += 256) {
      int rr  = i >> 5;        // local row
      int cc4 = i & 31;        // float4 column index
      int row = row0 + rr;
      if (row < M) {
        float4 x = t4[i];
        float4 b = b4[cc4];
        x.x += b.x; x.y += b.y; x.z += b.z; x.w += b.w;
        if (HAS_RES) {
          float4 r = r4[(long)row * (CH / 4) + cc4];
          x.x += r.x; x.y += r.y; x.z += r.z; x.w += r.w;
        }
        x.x = fmaxf(x.x, 0.f); x.y = fmaxf(x.y, 0.f);
        x.z = fmaxf(x.z, 0.f); x.w = fmaxf(x.w, 0.f);
        out4[(long)row * (CH / 4) + cc4] = x;
      }
    }
  }
}

// ---------------------------------------------------------------------------
// Host orchestration
// ---------------------------------------------------------------------------
extern "C" void kernel_launch(void* const* d_in, const int* in_sizes, int n_in,
                              void* d_out, int out_size, void* d_ws, size_t ws_size,
                              hipStream_t stream) {
  const float* hd      = (const float*)d_in[0];
  const float* hp      = (const float*)d_in[1];
  const int*   ei_d2p  = (const int*)d_in[2];
  const int*   ei_p2d  = (const int*)d_in[3];
  const int*   ei_p2p  = (const int*)d_in[4];
  const float* W1_dself = (const float*)d_in[5];
  const float* W1_p2d   = (const float*)d_in[6];
  const float* W1_pself = (const float*)d_in[7];
  const float* W1_d2p   = (const float*)d_in[8];
  const float* W1_p2p   = (const float*)d_in[9];
  const float* b1_d     = (const float*)d_in[10];
  const float* b1_p     = (const float*)d_in[11];
  const float* W2_dself = (const float*)d_in[12];
  const float* W2_p2d   = (const float*)d_in[13];
  const float* W2_pself = (const float*)d_in[14];
  const float* W2_d2p   = (const float*)d_in[15];
  const float* W2_p2p   = (const float*)d_in[16];
  const float* b2_d     = (const float*)d_in[17];
  const float* b2_p     = (const float*)d_in[18];

  const int ND = in_sizes[0] / CH;   // 50000
  const int NP = in_sizes[1] / CH;   // 50000
  const int E  = in_sizes[2] / 2;    // 640000

  float* out_d = (float*)d_out;
  float* out_p = out_d + (long)ND * CH;
  // Inter-layer activations live in d_out (in-place safe: each block fully
  // reads its A rows into LDS before its epilogue writes the same rows).
  float* h1d = out_d;
  float* h1p = out_p;

  // Workspace: three aggregation accumulators (76.8 MB).
  float* aggD  = (float*)d_ws;               // [ND x 128] <- prot via p2d
  float* aggP1 = aggD  + (long)ND * CH;      // [NP x 128] <- drug via d2p
  float* aggP2 = aggP1 + (long)NP * CH;      // [NP x 128] <- prot via p2p

  const long nAggF4   = ((long)ND + 2L * NP) * CH / 4;
  const int  zeroBlks = 4096;
  const int  scatBlks = (E * 32 + 255) / 256;   // one wave32 per edge
  const int  gD = (ND + 15) / 16;
  const int  gP = (NP + 15) / 16;

  // ---------------- Layer 1 ----------------
  zero_f4<<<zeroBlks, 256, 0, stream>>>((float4*)aggD, nAggF4);
  scatter_add<<<scatBlks, 256, 0, stream>>>(hp, ei_p2d, aggD,  E);
  scatter_add<<<scatBlks, 256, 0, stream>>>(hd, ei_d2p, aggP1, E);
  scatter_add<<<scatBlks, 256, 0, stream>>>(hp, ei_p2p, aggP2, E);
  gemm_multi<2, false><<<gD, 256, 0, stream>>>(hd, W1_dself, aggD, W1_p2d,
                                               nullptr, nullptr, b1_d, nullptr,
                                               h1d, ND);
  gemm_multi<3, false><<<gP, 256, 0, stream>>>(hp, W1_pself, aggP1, W1_d2p,
                                               aggP2, W1_p2p, b1_p, nullptr,
                                               h1p, NP);

  // ---------------- Layer 2 ----------------
  zero_f4<<<zeroBlks, 256, 0, stream>>>((float4*)aggD, nAggF4);
  scatter_add<<<scatBlks, 256, 0, stream>>>(h1p, ei_p2d, aggD,  E);
  scatter_add<<<scatBlks, 256, 0, stream>>>(h1d, ei_d2p, aggP1, E);
  scatter_add<<<scatBlks, 256, 0, stream>>>(h1p, ei_p2p, aggP2, E);
  gemm_multi<2, true><<<gD, 256, 0, stream>>>(h1d, W2_dself, aggD, W2_p2d,
                                              nullptr, nullptr, b2_d, hd,
                                              out_d, ND);
  gemm_multi<3, true><<<gP, 256, 0, stream>>>(h1p, W2_pself, aggP1, W2_d2p,
                                              aggP2, W2_p2p, b2_p, hp,
                                              out_p, NP);
}